// MATCHLoss_83726092468741
// MI455X (gfx1250) — compile-verified
//
#include <hip/hip_runtime.h>
#include <hip/hip_bf16.h>

#define N_LABELS 30000
#define HIDDEN   1024
#define BATCH    256
#define N_EDGES  (N_LABELS - 1)

// ws layout (floats):
//   [0   .. 255 ]  per-row BCE partial sums
//   [256 .. 511 ]  per-row prob_reg partial sums
//   [512 .. 4261]  per-block rec_reg partial sums (3750 blocks)
#define WS_BCE   0
#define WS_PROB  256
#define WS_REC   512
#define REC_BLOCKS ((N_EDGES + 7) / 8)

typedef __attribute__((ext_vector_type(4))) unsigned int uint32x4_t;
typedef __attribute__((ext_vector_type(8))) int          int32x8_t;
typedef __attribute__((ext_vector_type(4))) int          int32x4_t;

// ---------------------------------------------------------------------------
// Kernel 1: one block per batch row.
//   - TDM (tensor_load_to_lds) stages the 30000-float logits row into LDS
//     (120 KB -> fits CDNA5's 320 KB WGP LDS; 2 blocks/WGP occupancy).
//   - Fused elementwise pass: BCE partial + in-place sigmoid conversion.
//   - Edge loop: prob_reg gathers hit LDS instead of random global memory.
// ---------------------------------------------------------------------------
__global__ __launch_bounds__(256) void bce_prob_kernel(
    const float* __restrict__ logits,
    const float* __restrict__ targets,
    const int*   __restrict__ parent_idx,
    const int*   __restrict__ child_idx,
    float*       __restrict__ ws) {
  __shared__ float row[N_LABELS];  // 120000 bytes, LDS offset 0
  const int b   = blockIdx.x;
  const int tid = threadIdx.x;
  const float* __restrict__ lrow = logits + (size_t)b * N_LABELS;

#if defined(__gfx1250__) && __has_builtin(__builtin_amdgcn_tensor_load_to_lds)
  if (tid < 32) {  // a single wave issues the DMA descriptor
    unsigned long long ga = (unsigned long long)(size_t)lrow;
    uint32x4_t g0;
    int32x8_t  g1;
    int32x4_t  gz4 = {0, 0, 0, 0};
    int32x8_t  gz8 = {0, 0, 0, 0, 0, 0, 0, 0};
    // --- D# group 0: count=1, lds_addr=0 (row[] is the only LDS object),
    //     57-bit global tile address, type=2 ("image")
    g0.x = 1u;
    g0.y = 0u;
    g0.z = (unsigned)(ga & 0xffffffffu);
    g0.w = (unsigned)((ga >> 32) & 0x01ffffffu) | (2u << 30);
    // --- D# group 1: data_size=4B (code 2), tensor 30000 x 1, tile 30000 x 1,
    //     dim0 stride = 30000 elements, no padding/multicast/barrier.
    g1[0] = (int)(2u << 16);                                   // mask=0, data_size=2
    g1[1] = (int)(((unsigned)N_LABELS & 0xffffu) << 16);       // tensor_dim0 lo16
    g1[2] = (int)(((unsigned)N_LABELS >> 16) | (1u << 16));    // dim0 hi16 | tensor_dim1=1 lo
    g1[3] = (int)((unsigned)N_LABELS << 16);                   // dim1 hi=0 | tile_dim0=30000
    g1[4] = 1;                                                 // tile_dim1=1, tile_dim2=0
    g1[5] = (int)(unsigned)N_LABELS;                           // tensor_dim0_stride lo32
    g1[6] = (int)(((unsigned)N_LABELS & 0xffffu) << 16);       // stride hi=0 | dim1_stride lo
    g1[7] = (int)((unsigned)N_LABELS >> 16);                   // dim1_stride hi
    __builtin_amdgcn_tensor_load_to_lds(g0, g1, gz4, gz4, gz8, 0);
    __builtin_amdgcn_s_wait_tensorcnt(0);                      // TENSORcnt -> 0
  }
  __syncthreads();  // LDS row visible to all 8 waves
#else
  for (int i = tid; i < N_LABELS; i += 256) row[i] = lrow[i];
  __syncthreads();
#endif

  // Fused BCE + sigmoid conversion. One expf serves both:
  //   softplus(l) = max(l,0) + log1p(e),  e = exp(-|l|)
  //   sigmoid(l)  = l>=0 ? 1/(1+e) : e/(1+e)
  float bce = 0.0f;
  const float* __restrict__ trow = targets + (size_t)b * N_LABELS;
  for (int i = tid; i < N_LABELS; i += 256) {
    float l = row[i];
    float t = trow[i];
    float e = expf(-fabsf(l));
    float inv = 1.0f / (1.0f + e);
    bce += fmaxf(l, 0.0f) + log1pf(e) - l * t;
    row[i] = (l >= 0.0f) ? inv : e * inv;  // in-place: row now holds sigmoid
  }
  __syncthreads();

  // prob_reg: 29999 LDS gathers per thread-stride — stays on the WGP.
  float preg = 0.0f;
  for (int e = tid; e < N_EDGES; e += 256) {
    int p = parent_idx[e];
    int c = child_idx[e];
    float d = row[c] - row[p];
    preg += fmaxf(d, 0.0f);
  }
  __syncthreads();

  // Deterministic block tree-reduce (reuse first 512 LDS slots).
  row[tid]       = bce;
  row[256 + tid] = preg;
  __syncthreads();
  for (int s = 128; s > 0; s >>= 1) {
    if (tid < s) {
      row[tid]       += row[tid + s];
      row[256 + tid] += row[256 + tid + s];
    }
    __syncthreads();
  }
  if (tid == 0) {
    ws[WS_BCE  + b] = row[0];
    ws[WS_PROB + b] = row[256];
  }
}

// ---------------------------------------------------------------------------
// Kernel 2: rec_reg. One wave32 per edge (8 edges per 256-thread block).
// Two contiguous 4 KB rows per edge -> coalesced b128 loads, shfl reduce.
// ---------------------------------------------------------------------------
__global__ __launch_bounds__(256) void rec_kernel(
    const float* __restrict__ params,
    const int*   __restrict__ parent_idx,
    const int*   __restrict__ child_idx,
    float*       __restrict__ ws) {
  const int tid  = threadIdx.x;
  const int wave = tid >> 5;
  const int lane = tid & 31;
  const int edge = blockIdx.x * 8 + wave;
  const bool valid = (edge < N_EDGES);
  const int e = valid ? edge : (N_EDGES - 1);  // clamp; no early return (barriers below)

  const float4* __restrict__ pr =
      (const float4*)(params + (size_t)parent_idx[e] * HIDDEN);
  const float4* __restrict__ cr =
      (const float4*)(params + (size_t)child_idx[e] * HIDDEN);

  float s = 0.0f;
#pragma unroll
  for (int k = 0; k < HIDDEN / 128; ++k) {  // 8 x (32 lanes x float4) = 1024
    float4 a = pr[k * 32 + lane];
    float4 b = cr[k * 32 + lane];
    float d0 = a.x - b.x, d1 = a.y - b.y, d2 = a.z - b.z, d3 = a.w - b.w;
    s += d0 * d0 + d1 * d1 + d2 * d2 + d3 * d3;
  }
  if (!valid) s = 0.0f;

  // wave32 reduction
#pragma unroll
  for (int off = 16; off > 0; off >>= 1) s += __shfl_down(s, off, 32);

  __shared__ float red[8];
  if (lane == 0) red[wave] = s;
  __syncthreads();
  if (tid == 0) {
    float acc = 0.0f;
#pragma unroll
    for (int w = 0; w < 8; ++w) acc += red[w];
    ws[WS_REC + blockIdx.x] = acc;
  }
}

// ---------------------------------------------------------------------------
// Kernel 3: deterministic final combine of all partials.
// ---------------------------------------------------------------------------
__global__ __launch_bounds__(256) void combine_kernel(
    const float* __restrict__ ws, float* __restrict__ out) {
  const int tid = threadIdx.x;
  __shared__ float sb[256], sp[256], sr[256];
  float b = ws[WS_BCE + tid];
  float p = ws[WS_PROB + tid];
  float r = 0.0f;
  for (int i = tid; i < REC_BLOCKS; i += 256) r += ws[WS_REC + i];
  sb[tid] = b; sp[tid] = p; sr[tid] = r;
  __syncthreads();
  for (int s = 128; s > 0; s >>= 1) {
    if (tid < s) {
      sb[tid] += sb[tid + s];
      sp[tid] += sp[tid + s];
      sr[tid] += sr[tid + s];
    }
    __syncthreads();
  }
  if (tid == 0) {
    const float inv_mean = 1.0f / ((float)BATCH * (float)N_LABELS);
    out[0] = sb[0] * inv_mean + 1e-4f * (0.5f * sr[0]) + 1e-4f * sp[0];
  }
}

extern "C" void kernel_launch(void* const* d_in, const int* in_sizes, int n_in,
                              void* d_out, int out_size, void* d_ws, size_t ws_size,
                              hipStream_t stream) {
  const float* logits  = (const float*)d_in[0];
  const float* targets = (const float*)d_in[1];
  const float* params  = (const float*)d_in[2];
  const int*   parent  = (const int*)d_in[3];
  const int*   child   = (const int*)d_in[4];
  float* ws  = (float*)d_ws;   // needs (512 + 3750) * 4 B ~= 17 KB
  float* out = (float*)d_out;

  bce_prob_kernel<<<BATCH, 256, 0, stream>>>(logits, targets, parent, child, ws);
  rec_kernel<<<REC_BLOCKS, 256, 0, stream>>>(params, parent, child, ws);
  combine_kernel<<<1, 256, 0, stream>>>(ws, out);
}